// GraphSAGENEW_11081015623738
// MI455X (gfx1250) — compile-verified
//
#include <hip/hip_runtime.h>

typedef float v2f __attribute__((ext_vector_type(2)));
typedef float v8f __attribute__((ext_vector_type(8)));

#define NNODES 100000
#define NEDGES 1600000

// ---------------------------------------------------------------- utilities
__global__ void zero_f32(float* __restrict__ p, int n) {
  int i = blockIdx.x * blockDim.x + threadIdx.x;
  if (i < n) p[i] = 0.0f;
}

__global__ void degree_k(const long long* __restrict__ dst, float* __restrict__ deg) {
  int e = blockIdx.x * blockDim.x + threadIdx.x;
  if (e < NEDGES) atomicAdd(&deg[(int)dst[e]], 1.0f);
}

__global__ void invdeg_k(float* __restrict__ deg) {
  int i = blockIdx.x * blockDim.x + threadIdx.x;
  if (i < NNODES) {
    float d = deg[i];
    deg[i] = (d > 0.0f) ? (1.0f / d) : 0.0f;
  }
}

// ------------------------------------------------------- edge aggregation
// One thread per (edge, 4-channel group): float4 gather from src row,
// 4x global_atomic_add_f32 into dst row. Consecutive lanes cover the
// channels of one edge -> coalesced gather, L2-resident scatter.
template <int D>
__global__ void aggregate_k(const float* __restrict__ feat,
                            const long long* __restrict__ src,
                            const long long* __restrict__ dst,
                            float* __restrict__ agg) {
  constexpr int G = D / 4;  // 16 or 32, power of two
  long long gid = (long long)blockIdx.x * blockDim.x + threadIdx.x;
  if (gid >= (long long)NEDGES * G) return;
  int e = (int)(gid / G);
  int g = (int)(gid % G);
  int s = (int)src[e];
  int d = (int)dst[e];
  const float4 v = *(const float4*)(feat + (long long)s * D + g * 4);
  float* o = agg + (long long)d * D + g * 4;
  atomicAdd(o + 0, v.x);
  atomicAdd(o + 1, v.y);
  atomicAdd(o + 2, v.z);
  atomicAdd(o + 3, v.w);
}

// ------------------------------------------------- fused SAGE layer (WMMA)
// out[16x16 tile] = relu( (agg*inv_deg) @ Wl^T + b + x @ Wr^T )
// One wave per 16x16 tile, K stepped by 4 via V_WMMA_F32_16X16X4_F32.
// A layout (32-bit 16x4): lanes 0-15 -> M=lane, K={k,k+1}; lanes 16-31 ->
// M=lane-16, K={k+2,k+3}.  B[k][n] = W[n][k] mirrors it from row-major W.
// C/D: VGPR r, lane l -> M = r + 8*(l>=16), N = l&15.
template <int DI, int DO, bool RELU>
__global__ __launch_bounds__(128) void sage_layer_k(
    const float* __restrict__ agg, const float* __restrict__ x,
    const float* __restrict__ invdeg, const float* __restrict__ Wl,
    const float* __restrict__ bias, const float* __restrict__ Wr,
    float* __restrict__ out) {
  constexpr int MT = NNODES / 16;  // 6250 M-tiles, exact
  const int wave = threadIdx.x >> 5;
  const int lane = threadIdx.x & 31;
  const int mtile = blockIdx.x * 4 + wave;
  if (mtile >= MT) return;  // wave-uniform: EXEC stays all-ones for WMMA

  const int row0 = mtile * 16;
  const int col0 = blockIdx.y * 16;
  const int n = lane & 15;
  const int half = lane >> 4;
  const int ko = half * 2;  // K offset within the 4-step for this half-wave

  const float sA = invdeg[row0 + n];
  const float* aRow = agg + (long long)(row0 + n) * DI;
  const float* xRow = x + (long long)(row0 + n) * DI;
  const float* wlRow = Wl + (long long)(col0 + n) * DI;
  const float* wrRow = Wr + (long long)(col0 + n) * DI;

  v8f acc = {0.f, 0.f, 0.f, 0.f, 0.f, 0.f, 0.f, 0.f};

#pragma unroll
  for (int k = 0; k < DI; k += 4) {
    v2f a = *(const v2f*)(aRow + k + ko);
    a *= sA;  // mean-aggregate scaling fused into A fragment
    v2f b = *(const v2f*)(wlRow + k + ko);
    acc = __builtin_amdgcn_wmma_f32_16x16x4_f32(false, a, false, b, (short)0,
                                                acc, false, false);
  }
#pragma unroll
  for (int k = 0; k < DI; k += 4) {
    v2f a = *(const v2f*)(xRow + k + ko);
    v2f b = *(const v2f*)(wrRow + k + ko);
    acc = __builtin_amdgcn_wmma_f32_16x16x4_f32(false, a, false, b, (short)0,
                                                acc, false, false);
  }

  const float bv = bias[col0 + n];
#pragma unroll
  for (int r = 0; r < 8; ++r) {
    const int m = row0 + r + half * 8;
    float v = acc[r] + bv;
    if (RELU) v = fmaxf(v, 0.0f);
    out[(long long)m * DO + col0 + n] = v;
  }
}

// ----------------------------------------------------------------- driver
extern "C" void kernel_launch(void* const* d_in, const int* in_sizes, int n_in,
                              void* d_out, int out_size, void* d_ws,
                              size_t ws_size, hipStream_t stream) {
  (void)in_sizes; (void)n_in; (void)out_size; (void)ws_size;

  const float* x = (const float*)d_in[0];
  const long long* ei = (const long long*)d_in[1];
  const long long* src = ei;            // edge_index[0]
  const long long* dst = ei + NEDGES;   // edge_index[1]

  const float* Wl1 = (const float*)d_in[2];
  const float* b1  = (const float*)d_in[3];
  const float* Wr1 = (const float*)d_in[4];
  const float* Wl2 = (const float*)d_in[5];
  const float* b2  = (const float*)d_in[6];
  const float* Wr2 = (const float*)d_in[7];
  const float* Wl3 = (const float*)d_in[8];
  const float* b3  = (const float*)d_in[9];
  const float* Wr3 = (const float*)d_in[10];
  const float* Wl4 = (const float*)d_in[11];
  const float* b4  = (const float*)d_in[12];
  const float* Wr4 = (const float*)d_in[13];

  float* invdeg = (float*)d_ws;
  float* agg = invdeg + NNODES;
  float* hA = agg + (size_t)NNODES * 128;
  float* hB = hA + (size_t)NNODES * 128;
  float* outp = (float*)d_out;

  const int TB = 256;
  auto blk = [](long long n, int t) { return (unsigned)((n + t - 1) / t); };

  // degrees -> inverse degrees
  zero_f32<<<blk(NNODES, TB), TB, 0, stream>>>(invdeg, NNODES);
  degree_k<<<blk(NEDGES, TB), TB, 0, stream>>>(dst, invdeg);
  invdeg_k<<<blk(NNODES, TB), TB, 0, stream>>>(invdeg);

  const dim3 gemm_blk(128);
  const unsigned mblocks = (NNODES / 16 + 3) / 4;  // 4 waves/block

  // layer 1: 64 -> 128, relu
  zero_f32<<<blk((long long)NNODES * 64, TB), TB, 0, stream>>>(agg, NNODES * 64);
  aggregate_k<64><<<blk((long long)NEDGES * 16, TB), TB, 0, stream>>>(x, src, dst, agg);
  sage_layer_k<64, 128, true>
      <<<dim3(mblocks, 8), gemm_blk, 0, stream>>>(agg, x, invdeg, Wl1, b1, Wr1, hA);

  // layer 2: 128 -> 128, relu
  zero_f32<<<blk((long long)NNODES * 128, TB), TB, 0, stream>>>(agg, NNODES * 128);
  aggregate_k<128><<<blk((long long)NEDGES * 32, TB), TB, 0, stream>>>(hA, src, dst, agg);
  sage_layer_k<128, 128, true>
      <<<dim3(mblocks, 8), gemm_blk, 0, stream>>>(agg, hA, invdeg, Wl2, b2, Wr2, hB);

  // layer 3: 128 -> 128, relu
  zero_f32<<<blk((long long)NNODES * 128, TB), TB, 0, stream>>>(agg, NNODES * 128);
  aggregate_k<128><<<blk((long long)NEDGES * 32, TB), TB, 0, stream>>>(hB, src, dst, agg);
  sage_layer_k<128, 128, true>
      <<<dim3(mblocks, 8), gemm_blk, 0, stream>>>(agg, hB, invdeg, Wl3, b3, Wr3, hA);

  // layer 4: 128 -> 64, no relu, straight to d_out
  zero_f32<<<blk((long long)NNODES * 128, TB), TB, 0, stream>>>(agg, NNODES * 128);
  aggregate_k<128><<<blk((long long)NEDGES * 32, TB), TB, 0, stream>>>(hA, src, dst, agg);
  sage_layer_k<128, 64, false>
      <<<dim3(mblocks, 4), gemm_blk, 0, stream>>>(agg, hA, invdeg, Wl4, b4, Wr4, outp);
}